// PointProcessGPFA_56289841381389
// MI455X (gfx1250) — compile-verified
//
#include <hip/hip_runtime.h>
#include <math.h>

// Problem constants (match reference)
#define Tn    256
#define OBSn  48
#define LATn  8
#define Nn    2048           // LATn * Tn
#define NB    128            // Cholesky block size for Sigma
#define NSTEP (Nn / NB)      // 16

typedef float v2f __attribute__((ext_vector_type(2)));
typedef float v8f __attribute__((ext_vector_type(8)));

// FP32 WMMA: D(16x16) = A(16x4) * B(4x16) + C   -> v_wmma_f32_16x16x4_f32
__device__ __forceinline__ v8f wmma4(v2f a, v2f b, v8f c) {
  return __builtin_amdgcn_wmma_f32_16x16x4_f32(
      /*neg_a=*/false, a, /*neg_b=*/false, b,
      /*c_mod=*/(short)0, c, /*reuse_a=*/false, /*reuse_b=*/false);
}

// -------------------- small setup kernels --------------------

__global__ void init_scalars_kernel(float* scal) {
  if (threadIdx.x < 16) scal[threadIdx.x] = 0.0f;
}

// A8[l1,l2] = sum_i a[i] * W[i,l1] * W[i,l2]   (8x8)
__global__ void computeA_kernel(const float* __restrict__ W,
                                const float* __restrict__ a,
                                float* __restrict__ A8) {
  int t = threadIdx.x;
  if (t < 64) {
    int l1 = t & 7, l2 = t >> 3;
    float s = 0.0f;
    for (int i = 0; i < OBSn; ++i)
      s += a[i] * W[i * LATn + l1] * W[i * LATn + l2];
    A8[l2 * LATn + l1] = s;
  }
}

// Sigma base: 2*kron(A8, I_T)  (row = l1*T + t1)
__global__ void fillSigma_kernel(float* __restrict__ Sigma,
                                 const float* __restrict__ A8) {
  size_t idx = (size_t)blockIdx.x * 256 + threadIdx.x;
  int row = (int)(idx / Nn), col = (int)(idx % Nn);
  int l1 = row >> 8, t1 = row & 255;
  int l2 = col >> 8, t2 = col & 255;
  Sigma[idx] = (t1 == t2) ? 2.0f * A8[l1 * LATn + l2] : 0.0f;
}

// r[l*T+t] = sum_i W[i,l] * (y[t,i] - b[i])
__global__ void computeR_kernel(const float* __restrict__ y,
                                const float* __restrict__ W,
                                const float* __restrict__ b,
                                float* __restrict__ r) {
  int idx = blockIdx.x * 256 + threadIdx.x;
  if (idx < Nn) {
    int l = idx >> 8, t = idx & 255;
    float s = 0.0f;
    for (int i = 0; i < OBSn; ++i)
      s += W[i * LATn + l] * (y[t * OBSn + i] - b[i]);
    r[idx] = s;
  }
}

// -------------------- per-latent K block: Cholesky + inverse (in LDS) ---------
// One workgroup per latent. LDS: 256KB matrix + 16KB RHS panel (<=320KB/WGP).

__global__ __launch_bounds__(256, 1)
void kchol_kernel(const float* __restrict__ Kb,
                  float* __restrict__ Sigma,
                  float* __restrict__ scal) {
  __shared__ float Km[Tn * Tn];     // 256 KB
  __shared__ float Pm[Tn * 16];     // 16 KB
  __shared__ float sh_logdet;
  const int l = blockIdx.x, tid = threadIdx.x;
  const float* K0 = Kb + (size_t)l * Tn * Tn;

  for (int e = tid; e < Tn * Tn; e += 256) Km[e] = K0[e];
  if (tid == 0) sh_logdet = 0.0f;
  __syncthreads();

  // Right-looking Cholesky (lower), in LDS
  for (int j = 0; j < Tn; ++j) {
    if (tid == 0) {
      float d = sqrtf(Km[j * Tn + j]);
      Km[j * Tn + j] = d;
      sh_logdet += 2.0f * logf(d);
    }
    __syncthreads();
    float invd = 1.0f / Km[j * Tn + j];
    for (int i = j + 1 + tid; i < Tn; i += 256) Km[i * Tn + j] *= invd;
    __syncthreads();
    for (int i = j + 1 + tid; i < Tn; i += 256) {
      float lij = Km[i * Tn + j];
      for (int k = j + 1; k <= i; ++k) Km[i * Tn + k] -= lij * Km[k * Tn + j];
    }
    __syncthreads();
  }
  if (tid == 0) scal[8 + l] = sh_logdet;   // per-block slot: deterministic

  // Kinv = L^-T L^-1, solved 16 columns at a time; added into Sigma diag block.
  for (int cb = 0; cb < 16; ++cb) {
    int colbase = cb * 16;
    for (int e = tid; e < Tn * 16; e += 256) {
      int i = e >> 4, c = e & 15;
      Pm[e] = (i == colbase + c) ? 1.0f : 0.0f;
    }
    __syncthreads();
    // forward: L Z = E
    for (int i = 0; i < Tn; ++i) {
      if (tid < 16) Pm[i * 16 + tid] /= Km[i * Tn + i];
      __syncthreads();
      for (int kk = i + 1 + tid; kk < Tn; kk += 256) {
        float lk = Km[kk * Tn + i];
        #pragma unroll
        for (int c = 0; c < 16; ++c) Pm[kk * 16 + c] -= lk * Pm[i * 16 + c];
      }
      __syncthreads();
    }
    // backward: L^T X = Z
    for (int i = Tn - 1; i >= 0; --i) {
      if (tid < 16) Pm[i * 16 + tid] /= Km[i * Tn + i];
      __syncthreads();
      for (int kk = tid; kk < i; kk += 256) {
        float lk = Km[i * Tn + kk];   // L^T[kk][i]
        #pragma unroll
        for (int c = 0; c < 16; ++c) Pm[kk * 16 + c] -= lk * Pm[i * 16 + c];
      }
      __syncthreads();
    }
    // Sigma[l-block diag] += Kinv panel
    for (int e = tid; e < Tn * 16; e += 256) {
      int i = e >> 4, c = e & 15;
      size_t row = (size_t)(l * Tn + i), col = (size_t)(l * Tn + colbase + c);
      Sigma[row * Nn + col] += Pm[e];
    }
    __syncthreads();
  }
}

// -------------------- blocked Cholesky of Sigma (2048, NB=128) ----------------

// Factor diagonal block in LDS, compute its triangular inverse, write both out.
__global__ __launch_bounds__(256, 1)
void potrf_diag_kernel(float* __restrict__ Sigma,
                       float* __restrict__ Linv,
                       float* __restrict__ scal, int kstep) {
  __shared__ float Lb[NB * NB];   // 64 KB
  __shared__ float Xb[NB * NB];   // 64 KB
  __shared__ float sh_logdet;
  int tid = threadIdx.x;
  size_t base = (size_t)kstep * NB;

  for (int e = tid; e < NB * NB; e += 256)
    Lb[e] = Sigma[(base + (size_t)(e / NB)) * Nn + base + (e % NB)];
  if (tid == 0) sh_logdet = 0.0f;
  __syncthreads();

  for (int j = 0; j < NB; ++j) {
    if (tid == 0) {
      float d = sqrtf(Lb[j * NB + j]);
      Lb[j * NB + j] = d;
      sh_logdet += 2.0f * logf(d);
    }
    __syncthreads();
    float invd = 1.0f / Lb[j * NB + j];
    for (int i = j + 1 + tid; i < NB; i += 256) Lb[i * NB + j] *= invd;
    __syncthreads();
    for (int i = j + 1 + tid; i < NB; i += 256) {
      float lij = Lb[i * NB + j];
      for (int k = j + 1; k <= i; ++k) Lb[i * NB + k] -= lij * Lb[k * NB + j];
    }
    __syncthreads();
  }
  if (tid == 0) atomicAdd(&scal[0], sh_logdet);  // one WG per launch: deterministic

  // Xb = L^-1 (lower triangular inverse)
  for (int e = tid; e < NB * NB; e += 256)
    Xb[e] = ((e / NB) == (e % NB)) ? 1.0f : 0.0f;
  __syncthreads();
  for (int i = 0; i < NB; ++i) {
    float invd = 1.0f / Lb[i * NB + i];
    for (int c = tid; c <= i; c += 256) Xb[i * NB + c] *= invd;
    __syncthreads();
    int ncol = i + 1, tot = (NB - 1 - i) * ncol;
    for (int e = tid; e < tot; e += 256) {
      int kk = i + 1 + e / ncol, c = e % ncol;
      Xb[kk * NB + c] -= Lb[kk * NB + i] * Xb[i * NB + c];
    }
    __syncthreads();
  }
  for (int e = tid; e < NB * NB; e += 256) {
    Sigma[(base + (size_t)(e / NB)) * Nn + base + (e % NB)] = Lb[e];
    Linv[(size_t)kstep * NB * NB + e] = Xb[e];
  }
}

// TRSM as GEMM via fp32 WMMA: P = A21 * Linv^T.  One 16x16 tile per wave.
__global__ __launch_bounds__(32)
void trsm_kernel(const float* __restrict__ Sigma,
                 const float* __restrict__ Linv,
                 float* __restrict__ Pws, int kstep) {
  int lane = threadIdx.x;
  int m = lane & 15, h = lane >> 4;
  size_t rowg = (size_t)(kstep + 1) * NB + (size_t)blockIdx.x * 16;
  int colb = blockIdx.y * 16;
  const float* Lv = Linv + (size_t)kstep * NB * NB;
  size_t acol = (size_t)kstep * NB;

  v8f acc = {0.f, 0.f, 0.f, 0.f, 0.f, 0.f, 0.f, 0.f};
  for (int kk = 0; kk < NB; kk += 4) {
    v2f av, bv;
    // A[m][k]: vgpr v holds K = 2h+v (ISA 16x4 fp32 A layout)
    av.x = Sigma[(rowg + m) * Nn + acol + kk + 2 * h];
    av.y = Sigma[(rowg + m) * Nn + acol + kk + 2 * h + 1];
    // B[k][n] = Linv^T[k][n] = Linv[n][k], n = lane&15
    bv.x = Lv[(size_t)(colb + m) * NB + kk + 2 * h];
    bv.y = Lv[(size_t)(colb + m) * NB + kk + 2 * h + 1];
    acc = wmma4(av, bv, acc);
  }
  int rowt = blockIdx.x * 16;
  #pragma unroll
  for (int rr = 0; rr < 8; ++rr)   // D: vgpr rr -> row h*8+rr, col m
    Pws[(size_t)(rowt + h * 8 + rr) * NB + colb + m] = acc[rr];
}

// Copy TRSM panel back into Sigma's A21 region.
__global__ void copyback_kernel(const float* __restrict__ Pws,
                                float* __restrict__ Sigma, int kstep, int Mt) {
  int idx = blockIdx.x * 256 + threadIdx.x;
  if (idx < Mt * NB) {
    int i = idx / NB, j = idx % NB;
    Sigma[((size_t)(kstep + 1) * NB + i) * Nn + (size_t)kstep * NB + j] = Pws[idx];
  }
}

// SYRK trailing update via fp32 WMMA: A22 -= P * P^T (lower tiles only).
__global__ __launch_bounds__(32)
void syrk_kernel(float* __restrict__ Sigma,
                 const float* __restrict__ Pws, int kstep) {
  if (blockIdx.y > blockIdx.x) return;   // wave-uniform: EXEC stays all-ones
  int lane = threadIdx.x;
  int m = lane & 15, h = lane >> 4;
  int rt = blockIdx.x * 16, ct = blockIdx.y * 16;

  v8f acc = {0.f, 0.f, 0.f, 0.f, 0.f, 0.f, 0.f, 0.f};
  for (int kk = 0; kk < NB; kk += 4) {
    v2f av, bv;
    av.x = Pws[(size_t)(rt + m) * NB + kk + 2 * h];
    av.y = Pws[(size_t)(rt + m) * NB + kk + 2 * h + 1];
    bv.x = Pws[(size_t)(ct + m) * NB + kk + 2 * h];   // B = P^T
    bv.y = Pws[(size_t)(ct + m) * NB + kk + 2 * h + 1];
    acc = wmma4(av, bv, acc);
  }
  size_t tb = (size_t)(kstep + 1) * NB;
  #pragma unroll
  for (int rr = 0; rr < 8; ++rr) {
    size_t rg = tb + rt + h * 8 + rr;
    size_t cg = tb + ct + m;
    Sigma[rg * Nn + cg] -= acc[rr];
  }
}

// -------------------- blocked triangular solves + dot -------------------------

__global__ __launch_bounds__(256, 1)
void solve_kernel(const float* __restrict__ Sigma,
                  const float* __restrict__ Linv,
                  const float* __restrict__ r,
                  float* __restrict__ scal) {
  __shared__ float rsh[Nn], zsh[Nn], msh[Nn], tmp[NB], red[256];
  int tid = threadIdx.x;
  for (int e = tid; e < Nn; e += 256) rsh[e] = r[e];
  __syncthreads();

  // forward: L z = r
  for (int kb = 0; kb < NSTEP; ++kb) {
    if (tid < NB) {
      float acc = rsh[kb * NB + tid];
      for (int jb = 0; jb < kb; ++jb) {
        const float* Lrow = Sigma + (size_t)(kb * NB + tid) * Nn + (size_t)jb * NB;
        for (int x = 0; x < NB; ++x) acc -= Lrow[x] * zsh[jb * NB + x];
      }
      tmp[tid] = acc;
    }
    __syncthreads();
    if (tid < NB) {
      const float* Lv = Linv + (size_t)kb * NB * NB + (size_t)tid * NB;
      float s = 0.0f;
      for (int x = 0; x < NB; ++x) s += Lv[x] * tmp[x];
      zsh[kb * NB + tid] = s;
    }
    __syncthreads();
  }
  // backward: L^T mu = z
  for (int kb = NSTEP - 1; kb >= 0; --kb) {
    if (tid < NB) {
      float acc = zsh[kb * NB + tid];
      for (int jb = kb + 1; jb < NSTEP; ++jb)
        for (int x = 0; x < NB; ++x)
          acc -= Sigma[(size_t)(jb * NB + x) * Nn + (size_t)kb * NB + tid] *
                 msh[jb * NB + x];
      tmp[tid] = acc;
    }
    __syncthreads();
    if (tid < NB) {
      const float* Lv = Linv + (size_t)kb * NB * NB;
      float s = 0.0f;
      for (int x = 0; x < NB; ++x) s += Lv[(size_t)x * NB + tid] * tmp[x];  // Linv^T
      msh[kb * NB + tid] = s;
    }
    __syncthreads();
  }
  // dot(mu, r)
  float p = 0.0f;
  for (int e = tid; e < Nn; e += 256) p += msh[e] * rsh[e];
  red[tid] = p;
  __syncthreads();
  for (int s = 128; s > 0; s >>= 1) {
    if (tid < s) red[tid] += red[tid + s];
    __syncthreads();
  }
  if (tid == 0) scal[2] = red[0];
}

__global__ void finalize_kernel(const float* __restrict__ scal,
                                float* __restrict__ out) {
  if (threadIdx.x == 0 && blockIdx.x == 0) {
    float ldk = 0.0f;
    for (int l = 0; l < LATn; ++l) ldk += scal[8 + l];
    // 0.5*logdet(Sigma_inv) + 0.5*mu.r + 0.5*sum logdet(K_l)
    out[0] = 0.5f * (scal[0] + scal[2] + ldk);
  }
}

// -------------------- launch --------------------

extern "C" void kernel_launch(void* const* d_in, const int* in_sizes, int n_in,
                              void* d_out, int out_size, void* d_ws, size_t ws_size,
                              hipStream_t stream) {
  const float* y  = (const float*)d_in[0];   // [T, OBS]
  const float* W  = (const float*)d_in[1];   // [OBS, LAT]
  const float* Kb = (const float*)d_in[2];   // [LAT, T, T]
  const float* a  = (const float*)d_in[3];   // [OBS]
  const float* b  = (const float*)d_in[4];   // [OBS]
  float* out = (float*)d_out;

  float* ws    = (float*)d_ws;
  float* Sigma = ws;                                    // 2048*2048
  float* A8    = Sigma + (size_t)Nn * Nn;               // 64
  float* r     = A8 + 64;                               // 2048
  float* Linv  = r + Nn;                                // 16*128*128
  float* Pws   = Linv + (size_t)NSTEP * NB * NB;        // 1920*128
  float* scal  = Pws + (size_t)(Nn - NB) * NB;          // 16 scalars

  init_scalars_kernel<<<1, 32, 0, stream>>>(scal);
  computeA_kernel<<<1, 64, 0, stream>>>(W, a, A8);
  fillSigma_kernel<<<(unsigned)(((size_t)Nn * Nn) / 256), 256, 0, stream>>>(Sigma, A8);
  computeR_kernel<<<Nn / 256, 256, 0, stream>>>(y, W, b, r);
  kchol_kernel<<<LATn, 256, 0, stream>>>(Kb, Sigma, scal);

  for (int k = 0; k < NSTEP; ++k) {
    potrf_diag_kernel<<<1, 256, 0, stream>>>(Sigma, Linv, scal, k);
    int Mt = Nn - (k + 1) * NB;
    if (Mt > 0) {
      trsm_kernel<<<dim3(Mt / 16, NB / 16), 32, 0, stream>>>(Sigma, Linv, Pws, k);
      copyback_kernel<<<(Mt * NB + 255) / 256, 256, 0, stream>>>(Pws, Sigma, k, Mt);
      syrk_kernel<<<dim3(Mt / 16, Mt / 16), 32, 0, stream>>>(Sigma, Pws, k);
    }
  }

  solve_kernel<<<1, 256, 0, stream>>>(Sigma, Linv, r, scal);
  finalize_kernel<<<1, 32, 0, stream>>>(scal, out);
}